// GNNHybrid_2310692405933
// MI455X (gfx1250) — compile-verified
//
#include <hip/hip_runtime.h>
#include <math.h>

#define NIN 128
#define NH 128
#define NOUT 40
#define NOUTP 48

typedef __attribute__((ext_vector_type(2))) float v2f;
typedef __attribute__((ext_vector_type(8))) float v8f;

__device__ __forceinline__ v8f wmma4(v2f a, v2f b, v8f c) {
  // D = A(16x4,f32) * B(4x16,f32) + C(16x16,f32)
  return __builtin_amdgcn_wmma_f32_16x16x4_f32(false, a, false, b, (short)0, c,
                                               false, false);
}

__device__ __forceinline__ void atomAddF(float* p, float v) {
  unsafeAtomicAdd(p, v);  // HW global_atomic_add_f32
}

// Stage a 128x128 row-major weight into LDS with k-pair interleaving:
// Wp[(k>>1)*256 + c*2 + (k&1)] = W[k*128 + c]
// so that {W[k][c], W[k+1][c]} is one aligned 8-byte word -> single ds_load_b64.
__device__ __forceinline__ void stageW128(float* __restrict__ Wp,
                                          const float* __restrict__ W,
                                          int tid) {
  for (int i = tid; i < NH * NH; i += 256) {
    int k = i >> 7, c = i & 127;
    Wp[((k >> 1) << 8) + (c << 1) + (k & 1)] = W[i];
  }
}

// ---------------- edge aggregation: acc[dst] += feat[src]; optional degree ---
__global__ void k_edge_agg(const float* __restrict__ feat,
                           const int* __restrict__ src,
                           const int* __restrict__ dst,
                           float* __restrict__ acc, float* __restrict__ deg,
                           int nE, int withDeg) {
  unsigned t = blockIdx.x * blockDim.x + threadIdx.x;
  unsigned e = t >> 5;
  unsigned lane = t & 31u;
  if (e >= (unsigned)nE) return;
  int s = src[e], d = dst[e];
  const float4* xs = reinterpret_cast<const float4*>(feat + (size_t)s * NIN);
  float4 v = xs[lane];
  float* ad = acc + (size_t)d * NIN + lane * 4;
  atomAddF(ad + 0, v.x);
  atomAddF(ad + 1, v.y);
  atomAddF(ad + 2, v.z);
  atomAddF(ad + 3, v.w);
  if (withDeg && lane == 0) atomAddF(deg + d, 1.0f);
}

// ---------------- GEMM [N,128]x[128,128] + bias + LeakyReLU ------------------
__global__ __launch_bounds__(256) void k_gemm_leaky(
    const float* __restrict__ A, const float* __restrict__ W,
    const float* __restrict__ bias, float* __restrict__ Out, int nRows) {
  __shared__ float Ws[NH * NH];
  stageW128(Ws, W, threadIdx.x);
  __syncthreads();
  int wave = threadIdx.x >> 5, lane = threadIdx.x & 31;
  int ln = lane & 15, hl = lane >> 4;
  int m0 = (blockIdx.x * 8 + wave) * 16;
  if (m0 >= nRows) return;
  int row = m0 + ln;
  int rowc = row < nRows ? row : nRows - 1;
  const float* Ar = A + (size_t)rowc * NH;
  v8f acc[8];
#pragma unroll
  for (int nt = 0; nt < 8; ++nt)
#pragma unroll
    for (int r = 0; r < 8; ++r) acc[nt][r] = 0.0f;
  for (int k0 = 0; k0 < NH; k0 += 4) {
    v2f a = *(const v2f*)(Ar + k0 + 2 * hl);
    const float* Wrow = Ws + (((k0 >> 1) + hl) << 8) + (ln << 1);
#pragma unroll
    for (int nt = 0; nt < 8; ++nt) {
      v2f b = *(const v2f*)(Wrow + nt * 32);
      acc[nt] = wmma4(a, b, acc[nt]);
    }
  }
#pragma unroll
  for (int nt = 0; nt < 8; ++nt) {
    int c = nt * 16 + ln;
    float bv = bias[c];
#pragma unroll
    for (int r = 0; r < 8; ++r) {
      int m = m0 + r + 8 * hl;
      if (m < nRows) {
        float v = acc[nt][r] + bv;
        Out[(size_t)m * NH + c] = v > 0.0f ? v : 0.01f * v;
      }
    }
  }
}

// -------- GEMM + bias + row L2-normalize + BN-stat accumulation --------------
__global__ __launch_bounds__(256) void k_gemm_norm_stats(
    const float* __restrict__ A, const float* __restrict__ W,
    const float* __restrict__ bias, float* __restrict__ Out,
    float* __restrict__ statSum, float* __restrict__ statSq, int nRows) {
  __shared__ float Ws[NH * NH];
  stageW128(Ws, W, threadIdx.x);
  __syncthreads();
  int wave = threadIdx.x >> 5, lane = threadIdx.x & 31;
  int ln = lane & 15, hl = lane >> 4;
  int m0 = (blockIdx.x * 8 + wave) * 16;
  if (m0 >= nRows) return;
  int row = m0 + ln;
  int rowc = row < nRows ? row : nRows - 1;
  const float* Ar = A + (size_t)rowc * NH;
  v8f acc[8];
#pragma unroll
  for (int nt = 0; nt < 8; ++nt)
#pragma unroll
    for (int r = 0; r < 8; ++r) acc[nt][r] = 0.0f;
  for (int k0 = 0; k0 < NH; k0 += 4) {
    v2f a = *(const v2f*)(Ar + k0 + 2 * hl);
    const float* Wrow = Ws + (((k0 >> 1) + hl) << 8) + (ln << 1);
#pragma unroll
    for (int nt = 0; nt < 8; ++nt) {
      v2f b = *(const v2f*)(Wrow + nt * 32);
      acc[nt] = wmma4(a, b, acc[nt]);
    }
  }
#pragma unroll
  for (int nt = 0; nt < 8; ++nt) {
    float bv = bias[nt * 16 + ln];
#pragma unroll
    for (int r = 0; r < 8; ++r) acc[nt][r] += bv;
  }
  // per-row L2 norm: reduce across the 16-lane half (xor masks 1,2,4,8)
  float inv[8];
#pragma unroll
  for (int r = 0; r < 8; ++r) {
    float ss = 0.0f;
#pragma unroll
    for (int nt = 0; nt < 8; ++nt) ss += acc[nt][r] * acc[nt][r];
    ss += __shfl_xor(ss, 1, 32);
    ss += __shfl_xor(ss, 2, 32);
    ss += __shfl_xor(ss, 4, 32);
    ss += __shfl_xor(ss, 8, 32);
    inv[r] = 1.0f / fmaxf(sqrtf(ss), 1e-12f);
  }
#pragma unroll
  for (int nt = 0; nt < 8; ++nt) {
    int c = nt * 16 + ln;
    float s = 0.0f, sq = 0.0f;
#pragma unroll
    for (int r = 0; r < 8; ++r) {
      int m = m0 + r + 8 * hl;
      float v = (m < nRows) ? acc[nt][r] * inv[r] : 0.0f;
      if (m < nRows) Out[(size_t)m * NH + c] = v;
      s += v;
      sq += v * v;
    }
    s += __shfl_xor(s, 16, 32);
    sq += __shfl_xor(sq, 16, 32);
    if (hl == 0) {
      atomAddF(statSum + c, s);
      atomAddF(statSq + c, sq);
    }
  }
}

// -------- SAGE: (A1*invdeg)@Wl + A2@Wr + bias, + BN stats --------------------
__global__ __launch_bounds__(256) void k_gemm_sage(
    const float* __restrict__ A1, const float* __restrict__ A2,
    const float* __restrict__ Wl, const float* __restrict__ Wr,
    const float* __restrict__ bias, const float* __restrict__ deg,
    float* __restrict__ Out, float* __restrict__ statSum,
    float* __restrict__ statSq, int nRows) {
  __shared__ float Ws[NH * NH];
  int wave = threadIdx.x >> 5, lane = threadIdx.x & 31;
  int ln = lane & 15, hl = lane >> 4;
  int m0 = (blockIdx.x * 8 + wave) * 16;
  int row = m0 + ln;
  int rowc = row < nRows ? row : nRows - 1;
  v8f acc[8];
#pragma unroll
  for (int nt = 0; nt < 8; ++nt)
#pragma unroll
    for (int r = 0; r < 8; ++r) acc[nt][r] = 0.0f;
  // pass 1: mean-aggregated neighbours through Wl
  stageW128(Ws, Wl, threadIdx.x);
  __syncthreads();
  {
    float idg = 1.0f / fmaxf(deg[rowc], 1.0f);
    const float* Ar = A1 + (size_t)rowc * NH;
    for (int k0 = 0; k0 < NH; k0 += 4) {
      v2f a = *(const v2f*)(Ar + k0 + 2 * hl);
      a.x *= idg;
      a.y *= idg;
      const float* Wrow = Ws + (((k0 >> 1) + hl) << 8) + (ln << 1);
#pragma unroll
      for (int nt = 0; nt < 8; ++nt) {
        v2f b = *(const v2f*)(Wrow + nt * 32);
        acc[nt] = wmma4(a, b, acc[nt]);
      }
    }
  }
  __syncthreads();
  // pass 2: self features through Wr (accumulate into same C)
  stageW128(Ws, Wr, threadIdx.x);
  __syncthreads();
  {
    const float* Ar = A2 + (size_t)rowc * NH;
    for (int k0 = 0; k0 < NH; k0 += 4) {
      v2f a = *(const v2f*)(Ar + k0 + 2 * hl);
      const float* Wrow = Ws + (((k0 >> 1) + hl) << 8) + (ln << 1);
#pragma unroll
      for (int nt = 0; nt < 8; ++nt) {
        v2f b = *(const v2f*)(Wrow + nt * 32);
        acc[nt] = wmma4(a, b, acc[nt]);
      }
    }
  }
#pragma unroll
  for (int nt = 0; nt < 8; ++nt) {
    int c = nt * 16 + ln;
    float bv = bias[c];
    float s = 0.0f, sq = 0.0f;
#pragma unroll
    for (int r = 0; r < 8; ++r) {
      int m = m0 + r + 8 * hl;
      float v = (m < nRows) ? acc[nt][r] + bv : 0.0f;
      if (m < nRows) Out[(size_t)m * NH + c] = v;
      s += v;
      sq += v * v;
    }
    s += __shfl_xor(s, 16, 32);
    sq += __shfl_xor(sq, 16, 32);
    if (hl == 0) {
      atomAddF(statSum + c, s);
      atomAddF(statSq + c, sq);
    }
  }
}

// -------- GEMM [N,128]x[128,40] (padded to 48, k-pair interleaved LDS) -------
__global__ __launch_bounds__(256) void k_gemm_out(const float* __restrict__ A,
                                                  const float* __restrict__ W,
                                                  float* __restrict__ Out,
                                                  int nRows) {
  __shared__ float Ws[NH * NOUTP];  // pair layout: [(k>>1)*96 + c*2 + (k&1)]
  for (int i = threadIdx.x; i < NH * NOUTP; i += 256) {
    int k = i / NOUTP, c = i % NOUTP;
    float v = (c < NOUT) ? W[k * NOUT + c] : 0.0f;
    Ws[(k >> 1) * (2 * NOUTP) + (c << 1) + (k & 1)] = v;
  }
  __syncthreads();
  int wave = threadIdx.x >> 5, lane = threadIdx.x & 31;
  int ln = lane & 15, hl = lane >> 4;
  int m0 = (blockIdx.x * 8 + wave) * 16;
  if (m0 >= nRows) return;
  int row = m0 + ln;
  int rowc = row < nRows ? row : nRows - 1;
  const float* Ar = A + (size_t)rowc * NH;
  v8f acc[3];
#pragma unroll
  for (int nt = 0; nt < 3; ++nt)
#pragma unroll
    for (int r = 0; r < 8; ++r) acc[nt][r] = 0.0f;
  for (int k0 = 0; k0 < NH; k0 += 4) {
    v2f a = *(const v2f*)(Ar + k0 + 2 * hl);
    const float* Wrow = Ws + ((k0 >> 1) + hl) * (2 * NOUTP) + (ln << 1);
#pragma unroll
    for (int nt = 0; nt < 3; ++nt) {
      v2f b = *(const v2f*)(Wrow + nt * 32);
      acc[nt] = wmma4(a, b, acc[nt]);
    }
  }
#pragma unroll
  for (int nt = 0; nt < 3; ++nt) {
    int c = nt * 16 + ln;
    if (c < NOUT) {
#pragma unroll
      for (int r = 0; r < 8; ++r) {
        int m = m0 + r + 8 * hl;
        if (m < nRows) Out[(size_t)m * NOUT + c] = acc[nt][r];
      }
    }
  }
}

// ---------------- BN helpers -------------------------------------------------
__global__ void k_bn_finalize(const float* __restrict__ sum,
                              const float* __restrict__ sq,
                              const float* __restrict__ g,
                              const float* __restrict__ b,
                              float* __restrict__ scale,
                              float* __restrict__ shift, float invN) {
  int c = threadIdx.x;
  if (c < NH) {
    float mu = sum[c] * invN;
    float var = sq[c] * invN - mu * mu;
    float is = rsqrtf(var + 1e-5f);
    float sc = g[c] * is;
    scale[c] = sc;
    shift[c] = b[c] - mu * sc;
  }
}

__global__ void k_bn_relu(float* __restrict__ h, const float* __restrict__ scale,
                          const float* __restrict__ shift, int nElems) {
  int i = blockIdx.x * blockDim.x + threadIdx.x;
  if (i >= nElems) return;
  int c = i & (NH - 1);
  float v = h[i] * scale[c] + shift[c];
  h[i] = v > 0.0f ? v : 0.0f;
}

__global__ void k_dinv(const float* __restrict__ deg, float* __restrict__ dinv,
                       int n) {
  int i = blockIdx.x * blockDim.x + threadIdx.x;
  if (i < n) dinv[i] = rsqrtf(deg[i] + 1.0f);  // deg2 = deg + self-loop >= 1
}

// ---------------- GCN edge scatter: out[dst] += dinv[s]*dinv[d]*xw[src] ------
__global__ void k_gcn_edge(const float* __restrict__ xw,
                           const int* __restrict__ src,
                           const int* __restrict__ dst,
                           const float* __restrict__ dinv,
                           float* __restrict__ out, int nE) {
  unsigned t = blockIdx.x * blockDim.x + threadIdx.x;
  unsigned e = t / NOUT;
  unsigned c = t % NOUT;
  if (e >= (unsigned)nE) return;
  int s = src[e], d = dst[e];
  float w = dinv[s] * dinv[d];
  atomAddF(out + (size_t)d * NOUT + c, w * xw[(size_t)s * NOUT + c]);
}

// ---------------- self-loop + bias + log_softmax (one wave / row) ------------
__global__ void k_final(const float* __restrict__ accum,
                        const float* __restrict__ xw,
                        const float* __restrict__ dinv,
                        const float* __restrict__ bias, float* __restrict__ out,
                        int n) {
  int t = blockIdx.x * blockDim.x + threadIdx.x;
  int row = t >> 5;
  int lane = t & 31;
  if (row >= n) return;
  float di = dinv[row];
  float sl = di * di;
  size_t base = (size_t)row * NOUT;
  float v0 = accum[base + lane] + sl * xw[base + lane] + bias[lane];
  float v1 = -INFINITY;
  if (lane < NOUT - 32) {
    int c = lane + 32;
    v1 = accum[base + c] + sl * xw[base + c] + bias[c];
  }
  float m = fmaxf(v0, v1);
#pragma unroll
  for (int mk = 16; mk >= 1; mk >>= 1) m = fmaxf(m, __shfl_xor(m, mk, 32));
  float es = expf(v0 - m) + ((lane < NOUT - 32) ? expf(v1 - m) : 0.0f);
#pragma unroll
  for (int mk = 16; mk >= 1; mk >>= 1) es += __shfl_xor(es, mk, 32);
  float lse = m + logf(es);
  out[base + lane] = v0 - lse;
  if (lane < NOUT - 32) out[base + lane + 32] = v1 - lse;
}

// =============================================================================
extern "C" void kernel_launch(void* const* d_in, const int* in_sizes, int n_in,
                              void* d_out, int out_size, void* d_ws,
                              size_t ws_size, hipStream_t stream) {
  const float* x = (const float*)d_in[0];
  const int* edge = (const int*)d_in[1];
  const float* film_w1 = (const float*)d_in[2];
  const float* film_b1 = (const float*)d_in[3];
  const float* film_w2 = (const float*)d_in[4];
  const float* film_b2 = (const float*)d_in[5];
  const float* bn1_g = (const float*)d_in[6];
  const float* bn1_b = (const float*)d_in[7];
  const float* sage_wl = (const float*)d_in[8];
  const float* sage_bl = (const float*)d_in[9];
  const float* sage_wr = (const float*)d_in[10];
  const float* bn2_g = (const float*)d_in[11];
  const float* bn2_b = (const float*)d_in[12];
  const float* gcn_w = (const float*)d_in[13];
  const float* gcn_b = (const float*)d_in[14];

  const int N = in_sizes[0] / NIN;
  const int E = in_sizes[1] / 2;
  const int* src = edge;
  const int* dst = edge + E;

  float* ws = (float*)d_ws;
  size_t o = 0;
  float* buf0 = ws + o; o += (size_t)N * NH;   // neigh / neigh2 / xw
  float* buf1 = ws + o; o += (size_t)N * NH;   // h (layer1) / out-accum
  float* buf2 = ws + o; o += (size_t)N * NH;   // h1 / pre-BN2 / hh
  float* deg  = ws + o; o += (size_t)N;
  float* dinv = ws + o; o += (size_t)N;
  float* stats = ws + o;                        // 1024 floats
  float* sum1 = stats + 0,   *sq1 = stats + 128;
  float* sum2 = stats + 256, *sq2 = stats + 384;
  float* sc1  = stats + 512, *sh1 = stats + 640;
  float* sc2  = stats + 768, *sh2 = stats + 896;

  dim3 blk(256);
  unsigned aggBlocks  = ((unsigned)E * 32u + 255u) / 256u;
  unsigned gemmBlocks = (unsigned)((N + 127) / 128);
  unsigned bnBlocks   = ((unsigned)N * NH + 255u) / 256u;
  unsigned nBlocks    = ((unsigned)N + 255u) / 256u;
  unsigned geBlocks   = ((unsigned)E * NOUT + 255u) / 256u;
  unsigned fBlocks    = ((unsigned)N * 32u + 255u) / 256u;

  // ---- Layer 1: FiLM-ish ----
  hipMemsetAsync(buf0, 0, (size_t)N * NH * sizeof(float), stream);
  hipMemsetAsync(deg, 0, (size_t)N * sizeof(float), stream);
  hipMemsetAsync(stats, 0, 512 * sizeof(float), stream);
  k_edge_agg<<<aggBlocks, blk, 0, stream>>>(x, src, dst, buf0, deg, E, 1);
  k_gemm_leaky<<<gemmBlocks, blk, 0, stream>>>(buf0, film_w1, film_b1, buf2, N);
  k_gemm_norm_stats<<<gemmBlocks, blk, 0, stream>>>(buf2, film_w2, film_b2,
                                                    buf1, sum1, sq1, N);
  k_bn_finalize<<<1, 128, 0, stream>>>(sum1, sq1, bn1_g, bn1_b, sc1, sh1,
                                       1.0f / (float)N);
  k_bn_relu<<<bnBlocks, blk, 0, stream>>>(buf1, sc1, sh1, N * NH);

  // ---- Layer 2: SAGE (mean aggr) ----
  hipMemsetAsync(buf0, 0, (size_t)N * NH * sizeof(float), stream);
  k_edge_agg<<<aggBlocks, blk, 0, stream>>>(buf1, src, dst, buf0, deg, E, 0);
  k_gemm_sage<<<gemmBlocks, blk, 0, stream>>>(buf0, buf1, sage_wl, sage_wr,
                                              sage_bl, deg, buf2, sum2, sq2, N);
  k_bn_finalize<<<1, 128, 0, stream>>>(sum2, sq2, bn2_g, bn2_b, sc2, sh2,
                                       1.0f / (float)N);
  k_bn_relu<<<bnBlocks, blk, 0, stream>>>(buf2, sc2, sh2, N * NH);

  // ---- Layer 3: GCN + log_softmax ----
  k_dinv<<<nBlocks, blk, 0, stream>>>(deg, dinv, N);
  k_gemm_out<<<gemmBlocks, blk, 0, stream>>>(buf2, gcn_w, buf0, N);
  hipMemsetAsync(buf1, 0, (size_t)N * NOUT * sizeof(float), stream);
  k_gcn_edge<<<geBlocks, blk, 0, stream>>>(buf0, src, dst, dinv, buf1, E);
  k_final<<<fBlocks, blk, 0, stream>>>(buf1, buf0, dinv, gcn_b, (float*)d_out,
                                       N);
}